// JointSupervisedGroundedCoreferencer_60387240182304
// MI455X (gfx1250) — compile-verified
//
#include <hip/hip_runtime.h>
#include <hip/hip_bf16.h>

// ---------------------------------------------------------------------------
// Problem constants (from reference): B=128, N=32, L=36, Ds=512, Dv=1024, H=512
// ---------------------------------------------------------------------------
#define B_    128
#define N_    32
#define L_    36
#define DS_   512
#define DV_   1024
#define H_    512
#define NPAIR 496          // C(32,2)
#define K1    1536         // 3*Ds
#define SPAN_STRIDE 516    // padded LDS stride (breaks 64-bank conflicts)
#define H1T_STRIDE  36     // h1 stored transposed [col][row], padded

typedef __attribute__((ext_vector_type(2))) float v2f;
typedef __attribute__((ext_vector_type(8))) float v8f;

// CDNA5 f32 WMMA: D(16x16,f32) = A(16x4,f32) * B(4x16,f32) + C
__device__ __forceinline__ v8f wmma_f32_16x16x4(v2f a, v2f b, v8f c) {
  // 8 args: (neg_a, A, neg_b, B, c_mod, C, reuse_a, reuse_b)
  return __builtin_amdgcn_wmma_f32_16x16x4_f32(false, a, false, b, (short)0, c,
                                               false, false);
}

// ---------------------------------------------------------------------------
// Kernel A: G[b] = span[b] (32x512) @ Wg (512x1024)   -> ws
// grid (4 col-chunks of 256, 128 batches), 256 threads = 8 waves.
// Each wave: 2 row-tiles x 2 col-tiles of 16x16, K-loop step 4.
// ---------------------------------------------------------------------------
__global__ __launch_bounds__(256) void k_gemm_G(const float* __restrict__ span,
                                                const float* __restrict__ Wg,
                                                float* __restrict__ G) {
  extern __shared__ float smem[];
  float* spanS = smem;  // 32 x SPAN_STRIDE

  const int b     = blockIdx.y;
  const int chunk = blockIdx.x;           // 256 cols per chunk
  const int tid   = threadIdx.x;
  const int wave  = tid >> 5;
  const int lane  = tid & 31;
  const int half  = lane >> 4;
  const int lm    = lane & 15;

  for (int idx = tid; idx < N_ * DS_; idx += 256) {
    int r = idx >> 9, k = idx & 511;
    spanS[r * SPAN_STRIDE + k] = span[(b * N_ + r) * DS_ + k];
  }
  __syncthreads();

  v8f zero = {};
  v8f acc[2][2];
  for (int rt = 0; rt < 2; ++rt)
    for (int c = 0; c < 2; ++c) acc[rt][c] = zero;

  for (int k0 = 0; k0 < DS_; k0 += 4) {
    const int ka = k0 + half * 2;
    v2f a0 = *(const v2f*)&spanS[(lm) * SPAN_STRIDE + ka];
    v2f a1 = *(const v2f*)&spanS[(16 + lm) * SPAN_STRIDE + ka];
#pragma unroll
    for (int c = 0; c < 2; ++c) {
      const int col = chunk * 256 + wave * 32 + c * 16 + lm;
      v2f bb;
      bb.x = Wg[ka * DV_ + col];
      bb.y = Wg[(ka + 1) * DV_ + col];
      acc[0][c] = wmma_f32_16x16x4(a0, bb, acc[0][c]);
      acc[1][c] = wmma_f32_16x16x4(a1, bb, acc[1][c]);
    }
  }

#pragma unroll
  for (int rt = 0; rt < 2; ++rt)
#pragma unroll
    for (int c = 0; c < 2; ++c) {
      const int col = chunk * 256 + wave * 32 + c * 16 + lm;
#pragma unroll
      for (int r = 0; r < 8; ++r) {
        const int row = rt * 16 + half * 8 + r;
        G[(b * N_ + row) * DV_ + col] = acc[rt][c][r];
      }
    }
}

// ---------------------------------------------------------------------------
// Kernel B: M[b,n,l] = dot(G[b,n,:], image[b,l,:])   (0.3 GFLOP, L2-resident)
// One wave per output element, lane-strided reduction.
// ---------------------------------------------------------------------------
__global__ __launch_bounds__(256) void k_M(const float* __restrict__ G,
                                           const float* __restrict__ img,
                                           float* __restrict__ Mout) {
  const int wid  = blockIdx.x * (blockDim.x >> 5) + (threadIdx.x >> 5);
  const int lane = threadIdx.x & 31;
  if (wid >= B_ * N_ * L_) return;
  const int b   = wid / (N_ * L_);
  const int rem = wid - b * (N_ * L_);
  const int n   = rem / L_;
  const int l   = rem - n * L_;
  const float* g = G + (size_t)(b * N_ + n) * DV_;
  const float* v = img + (size_t)(b * L_ + l) * DV_;
  float s = 0.f;
  for (int k = lane; k < DV_; k += 32) s += g[k] * v[k];
#pragma unroll
  for (int off = 16; off > 0; off >>= 1) s += __shfl_xor(s, off, 32);
  if (lane == 0) Mout[wid] = s;
}

// ---------------------------------------------------------------------------
// Kernel C: closed-form masked softmax ->
//   sent[i,j] = (sum_{n,l} sm[i,n] im[j,l] M[i,n,l]) / (cnt_i * cnt_j)
//   (uniform mean of M[i] if either mask row is all-zero: softmax of all -inf)
// One block per i (128 threads = j).
// ---------------------------------------------------------------------------
__global__ __launch_bounds__(128) void k_sent(const float* __restrict__ M,
                                              const float* __restrict__ sm,
                                              const float* __restrict__ imk,
                                              float* __restrict__ sent) {
  __shared__ float S[L_];
  __shared__ float raw[L_];
  __shared__ float cntI_s, meanAll_s;
  const int i = blockIdx.x;
  const int t = threadIdx.x;

  if (t < L_) {
    float s = 0.f, rw = 0.f;
    for (int n = 0; n < N_; ++n) {
      float m = M[(i * N_ + n) * L_ + t];
      rw += m;
      s += sm[i * N_ + n] * m;
    }
    S[t] = s;
    raw[t] = rw;
  }
  if (t == 0) {
    float c = 0.f;
    for (int n = 0; n < N_; ++n) c += sm[i * N_ + n];
    cntI_s = c;
  }
  __syncthreads();
  if (t == 0) {
    float a = 0.f;
    for (int l = 0; l < L_; ++l) a += raw[l];
    meanAll_s = a / (float)(N_ * L_);
  }
  __syncthreads();

  float cntJ = 0.f, dot = 0.f;
  for (int l = 0; l < L_; ++l) {
    float w = imk[t * L_ + l];
    cntJ += w;
    dot += w * S[l];
  }
  const float cntI = cntI_s;
  sent[i * B_ + t] =
      (cntI > 0.f && cntJ > 0.f) ? dot / (cntI * cntJ) : meanAll_s;
}

// ---------------------------------------------------------------------------
// Kernel D: loss = ( -sum log_softmax(sent, rows) - sum log_softmax(sent^T) )/B
// Single block; thread i handles row i and column i.
// ---------------------------------------------------------------------------
__global__ __launch_bounds__(128) void k_loss(const float* __restrict__ sent,
                                              float* __restrict__ loss) {
  __shared__ float red[128];
  const int i = threadIdx.x;

  float mx = -3.4e38f;
  for (int j = 0; j < B_; ++j) mx = fmaxf(mx, sent[i * B_ + j]);
  float z = 0.f, rs = 0.f;
  for (int j = 0; j < B_; ++j) {
    float v = sent[i * B_ + j];
    z += expf(v - mx);
    rs += v;
  }
  float term = (float)B_ * (mx + logf(z)) - rs;

  mx = -3.4e38f;
  for (int j = 0; j < B_; ++j) mx = fmaxf(mx, sent[j * B_ + i]);
  z = 0.f;
  rs = 0.f;
  for (int j = 0; j < B_; ++j) {
    float v = sent[j * B_ + i];
    z += expf(v - mx);
    rs += v;
  }
  term += (float)B_ * (mx + logf(z)) - rs;

  red[i] = term;
  __syncthreads();
  for (int off = 64; off > 0; off >>= 1) {
    if (i < off) red[i] += red[i + off];
    __syncthreads();
  }
  if (i == 0) loss[0] = red[0] / (float)B_;
}

// ---------------------------------------------------------------------------
// Kernel E: fused pair-MLP (the 133 GFLOP hot path).
// grid (16 pair-tiles of 32, 128 batches), 256 threads = 8 waves.
// feat rows synthesized from LDS span[b]; h1 kept transposed in LDS; W3 dot
// reduced via shfl_xor + LDS atomics. All GEMMs via v_wmma_f32_16x16x4_f32.
// ---------------------------------------------------------------------------
__global__ __launch_bounds__(256) void k_mlp(
    const float* __restrict__ span, const float* __restrict__ W1,
    const float* __restrict__ b1, const float* __restrict__ W2,
    const float* __restrict__ b2, const float* __restrict__ W3,
    const float* __restrict__ b3, float* __restrict__ ts) {
  extern __shared__ float smem[];
  float* spanS = smem;                       // 32 x SPAN_STRIDE
  float* h1T   = spanS + N_ * SPAN_STRIDE;   // H_ x H1T_STRIDE  ([col][row])
  int*   iuS   = (int*)(h1T + H_ * H1T_STRIDE);
  int*   juS   = iuS + 32;
  float* tsS   = (float*)(juS + 32);         // 32

  const int b    = blockIdx.y;
  const int p0   = blockIdx.x * 32;
  const int tid  = threadIdx.x;
  const int wave = tid >> 5;
  const int lane = tid & 31;
  const int half = lane >> 4;
  const int lm   = lane & 15;

  if (tid < 32) {
    tsS[tid] = 0.f;
    int p = p0 + tid;
    if (p >= NPAIR) p = 0;  // dummy pair for padding rows (outputs masked)
    int i = 0, base = 0;
    while (base + (N_ - 1 - i) <= p) { base += N_ - 1 - i; ++i; }
    iuS[tid] = i;
    juS[tid] = i + 1 + (p - base);
  }
  for (int idx = tid; idx < N_ * DS_; idx += 256) {
    int r = idx >> 9, k = idx & 511;
    spanS[r * SPAN_STRIDE + k] = span[(b * N_ + r) * DS_ + k];
  }
  __syncthreads();

  v8f zero = {};

  // ---- layer 1: (32 x 1536) @ W1(1536 x 512), feat built on the fly ----
  v8f acc[2][4];
  for (int rt = 0; rt < 2; ++rt)
    for (int c = 0; c < 4; ++c) acc[rt][c] = zero;

  for (int k0 = 0; k0 < K1; k0 += 4) {
    const int ka = k0 + half * 2;
    v2f afr[2];
#pragma unroll
    for (int rt = 0; rt < 2; ++rt) {
      const int prow = rt * 16 + lm;
      const int iu = iuS[prow], ju = juS[prow];
      float x0, x1;
      if (ka < DS_) {                 // "first"
        x0 = spanS[iu * SPAN_STRIDE + ka];
        x1 = spanS[iu * SPAN_STRIDE + ka + 1];
      } else if (ka < 2 * DS_) {      // "second"
        x0 = spanS[ju * SPAN_STRIDE + ka - DS_];
        x1 = spanS[ju * SPAN_STRIDE + ka - DS_ + 1];
      } else {                        // "first * second"
        const int kk = ka - 2 * DS_;
        x0 = spanS[iu * SPAN_STRIDE + kk] * spanS[ju * SPAN_STRIDE + kk];
        x1 = spanS[iu * SPAN_STRIDE + kk + 1] * spanS[ju * SPAN_STRIDE + kk + 1];
      }
      afr[rt].x = x0;
      afr[rt].y = x1;
    }
#pragma unroll
    for (int c = 0; c < 4; ++c) {
      const int col = wave * 64 + c * 16 + lm;
      v2f bb;
      bb.x = W1[ka * H_ + col];
      bb.y = W1[(ka + 1) * H_ + col];
      acc[0][c] = wmma_f32_16x16x4(afr[0], bb, acc[0][c]);
      acc[1][c] = wmma_f32_16x16x4(afr[1], bb, acc[1][c]);
    }
  }

  // bias + relu -> h1T (transposed so layer-2 A-reads are conflict-light)
#pragma unroll
  for (int rt = 0; rt < 2; ++rt)
#pragma unroll
    for (int c = 0; c < 4; ++c) {
      const int col = wave * 64 + c * 16 + lm;
      const float bv = b1[col];
#pragma unroll
      for (int r = 0; r < 8; ++r) {
        const int row = rt * 16 + half * 8 + r;
        h1T[col * H1T_STRIDE + row] = fmaxf(acc[rt][c][r] + bv, 0.f);
      }
    }
  __syncthreads();

  // ---- layer 2: (32 x 512) @ W2(512 x 512) ----
  v8f acc2[2][4];
  for (int rt = 0; rt < 2; ++rt)
    for (int c = 0; c < 4; ++c) acc2[rt][c] = zero;

  for (int k0 = 0; k0 < H_; k0 += 4) {
    const int ka = k0 + half * 2;
    v2f afr[2];
#pragma unroll
    for (int rt = 0; rt < 2; ++rt) {
      const int row = rt * 16 + lm;
      afr[rt].x = h1T[ka * H1T_STRIDE + row];
      afr[rt].y = h1T[(ka + 1) * H1T_STRIDE + row];
    }
#pragma unroll
    for (int c = 0; c < 4; ++c) {
      const int col = wave * 64 + c * 16 + lm;
      v2f bb;
      bb.x = W2[ka * H_ + col];
      bb.y = W2[(ka + 1) * H_ + col];
      acc2[0][c] = wmma_f32_16x16x4(afr[0], bb, acc2[0][c]);
      acc2[1][c] = wmma_f32_16x16x4(afr[1], bb, acc2[1][c]);
    }
  }

  // bias + relu, dot with W3, reduce across lanes then waves
  float rs[16];
#pragma unroll
  for (int q = 0; q < 16; ++q) rs[q] = 0.f;
#pragma unroll
  for (int rt = 0; rt < 2; ++rt)
#pragma unroll
    for (int c = 0; c < 4; ++c) {
      const int col = wave * 64 + c * 16 + lm;
      const float bv = b2[col];
      const float w3 = W3[col];
#pragma unroll
      for (int r = 0; r < 8; ++r) {
        rs[rt * 8 + r] += fmaxf(acc2[rt][c][r] + bv, 0.f) * w3;
      }
    }
#pragma unroll
  for (int q = 0; q < 16; ++q) {
#pragma unroll
    for (int m = 8; m > 0; m >>= 1) rs[q] += __shfl_xor(rs[q], m, 16);
  }
  if (lm == 0) {
#pragma unroll
    for (int rt = 0; rt < 2; ++rt)
#pragma unroll
      for (int r = 0; r < 8; ++r) {
        const int row = rt * 16 + half * 8 + r;
        atomicAdd(&tsS[row], rs[rt * 8 + r]);  // ds_add_f32
      }
  }
  __syncthreads();
  if (tid < 32) {
    const int p = p0 + tid;
    if (p < NPAIR) ts[b * NPAIR + p] = tsS[tid] + b3[0];
  }
}

// ---------------------------------------------------------------------------
// Launch
// ---------------------------------------------------------------------------
extern "C" void kernel_launch(void* const* d_in, const int* in_sizes, int n_in,
                              void* d_out, int out_size, void* d_ws,
                              size_t ws_size, hipStream_t stream) {
  const float* span = (const float*)d_in[0];
  const float* img  = (const float*)d_in[1];
  const float* sm   = (const float*)d_in[2];
  const float* im   = (const float*)d_in[3];
  const float* Wg   = (const float*)d_in[4];
  const float* W1   = (const float*)d_in[5];
  const float* b1   = (const float*)d_in[6];
  const float* W2   = (const float*)d_in[7];
  const float* b2   = (const float*)d_in[8];
  const float* W3   = (const float*)d_in[9];
  const float* b3   = (const float*)d_in[10];

  float* out    = (float*)d_out;
  float* M_out  = out + 1;                 // (128,32,36)
  float* ts_out = out + 1 + B_ * N_ * L_;  // (128,496)

  float* G    = (float*)d_ws;              // 128*32*1024 floats = 16 MB
  float* sent = G + (size_t)B_ * N_ * DV_; // 128*128 floats

  const size_t shA = (size_t)N_ * SPAN_STRIDE * sizeof(float);
  k_gemm_G<<<dim3(4, B_), 256, shA, stream>>>(span, Wg, G);
  k_M<<<(B_ * N_ * L_) / 8, 256, 0, stream>>>(G, img, M_out);
  k_sent<<<B_, 128, 0, stream>>>(M_out, sm, im, sent);
  k_loss<<<1, 128, 0, stream>>>(sent, out);

  const size_t shE = shA + (size_t)H_ * H1T_STRIDE * sizeof(float) +
                     64 * sizeof(int) + 32 * sizeof(float);
  k_mlp<<<dim3((NPAIR + 31) / 32, B_), 256, shE, stream>>>(span, W1, b1, W2,
                                                           b2, W3, b3, ts_out);
}